// trNMS_43791486550556
// MI455X (gfx1250) — compile-verified
//
#include <hip/hip_runtime.h>
#include <stdint.h>

#define PANELTY_K       0.055f
#define HANNING_WEIGHT  0.42f
#define PI_F            3.14159265358979f

constexpr int NP_MAX = 2000;   // proposals per roi (reference: NP)
constexpr int BLK    = 256;    // 8 wave32 per block

__global__ __launch_bounds__(BLK)
void trnms_kernel(const float* __restrict__ rois,      // (N,4)
                  const float* __restrict__ rpn,       // (N,NP,5)
                  const float* __restrict__ sco,       // (N,NP,2)
                  float* __restrict__ out,             // N*4 coords then N scores
                  int N, int NP)
{
    __shared__ float s_rpn[NP_MAX * 5];   // 40000 B
    __shared__ float s_sc [NP_MAX * 2];   // 16000 B
    __shared__ float s_val[BLK];
    __shared__ int   s_idx[BLK];

    const int i   = blockIdx.x;
    const int tid = threadIdx.x;

    const float* rpn_row = rpn + (size_t)i * NP * 5;
    const float* sc_row  = sco + (size_t)i * NP * 2;

    // ---- Phase 1: async-stage both rows into LDS (CDNA5 async path, ASYNCcnt) ----
    // Low 32 bits of the flat LDS pointer are the group-segment byte offset.
    const uint32_t lds_rpn = (uint32_t)(size_t)(void*)s_rpn;
    const uint32_t lds_sc  = (uint32_t)(size_t)(void*)s_sc;
    const uint64_t rpn_base = (uint64_t)(size_t)(const void*)rpn_row;
    const uint64_t sc_base  = (uint64_t)(size_t)(const void*)sc_row;

    const int n_rpn16 = (NP * 5) / 4;     // b128 chunks (2500 for NP=2000)
    const int n_sc16  = (NP * 2) / 4;     // b128 chunks (1000)

    for (int t = tid; t < n_rpn16; t += BLK) {
        uint32_t go = (uint32_t)t * 16u;          // global byte offset within row
        uint32_t lo = lds_rpn + go;               // LDS byte address
        asm volatile("global_load_async_to_lds_b128 %0, %1, %2"
                     :: "v"(lo), "v"(go), "s"(rpn_base) : "memory");
    }
    for (int t = tid; t < n_sc16; t += BLK) {
        uint32_t go = (uint32_t)t * 16u;
        uint32_t lo = lds_sc + go;
        asm volatile("global_load_async_to_lds_b128 %0, %1, %2"
                     :: "v"(lo), "v"(go), "s"(sc_base) : "memory");
    }
    // tails (NP not multiple of 4) via plain copies
    for (int d = n_rpn16 * 4 + tid; d < NP * 5; d += BLK) s_rpn[d] = rpn_row[d];
    for (int d = n_sc16  * 4 + tid; d < NP * 2; d += BLK) s_sc[d]  = sc_row[d];

    asm volatile("s_wait_asynccnt 0" ::: "memory");
    __syncthreads();

    // ---- Reference-roi scalars (uniform per block; scalar loads) ----
    const float* r0 = rois + (size_t)i * 4;
    const float rx1 = r0[0], ry1 = r0[1], rx2 = r0[2], ry2 = r0[3];
    const float X  = (rx1 + rx2) * 0.5f;
    const float Y  = (ry1 + ry2) * 0.5f;
    const float Wd = fabsf(rx1 - rx2) + 0.0001f;
    const float Hd = fabsf(ry1 - ry2) + 0.0001f;
    const float P  = (Wd + Hd) * 0.5f;
    const float S  = __builtin_amdgcn_sqrtf((Wd + P) * (Hd + P));
    const float R  = Wd * __builtin_amdgcn_rcpf(Hd);
    const float Rmax = fmaxf(R, __builtin_amdgcn_rcpf(R));
    const float win  = S * 2.0f;                          // HANNING_WINDOW_SIZE_FACTOR
    const float wscl = PI_F * __builtin_amdgcn_rcpf(win); // pi / window_sz
    const float invS = __builtin_amdgcn_rcpf(S);

    // ---- Phase 2: score proposals out of LDS, per-thread argmax ----
    float best = -INFINITY;
    int   bidx = 0;
    for (int j = tid; j < NP; j += BLK) {
        const float x1 = s_rpn[j * 5 + 1];
        const float y1 = s_rpn[j * 5 + 2];
        const float x2 = s_rpn[j * 5 + 3];
        const float y2 = s_rpn[j * 5 + 4];
        const float sc1 = s_sc[j * 2 + 1];

        const float w  = fabsf(x1 - x2) + 0.0001f;
        const float h  = fabsf(y1 - y2) + 0.0001f;
        const float p  = (w + h) * 0.5f;
        const float s_ = __builtin_amdgcn_sqrtf((w + p) * (h + p));

        const float smax = fmaxf(S * __builtin_amdgcn_rcpf(s_), s_ * invS);
        const float pen  = __expf(-PANELTY_K * (smax * Rmax - 1.0f));

        const float dx = X - (x1 + x2) * 0.5f;
        const float dy = Y - (y1 + y2) * 0.5f;
        const float dist = __builtin_amdgcn_sqrtf(dx * dx + dy * dy);
        float han = 0.5f + 0.5f * __cosf(dist * wscl);
        han = (dist > win) ? 0.0f : han;

        const float val = sc1 * pen + han * HANNING_WEIGHT;
        if (val > best) { best = val; bidx = j; }   // strict '>' keeps first max
    }

    // ---- Phase 3: block argmax reduction (min index breaks ties) ----
    s_val[tid] = best;
    s_idx[tid] = bidx;
    __syncthreads();
    for (int off = BLK / 2; off > 0; off >>= 1) {
        if (tid < off) {
            const float v1 = s_val[tid],        v2 = s_val[tid + off];
            const int   i1 = s_idx[tid],        i2 = s_idx[tid + off];
            if (v2 > v1 || (v2 == v1 && i2 < i1)) { s_val[tid] = v2; s_idx[tid] = i2; }
        }
        __syncthreads();
    }

    if (tid == 0) {
        const int b = s_idx[0];
        out[(size_t)i * 4 + 0] = s_rpn[b * 5 + 1];
        out[(size_t)i * 4 + 1] = s_rpn[b * 5 + 2];
        out[(size_t)i * 4 + 2] = s_rpn[b * 5 + 3];
        out[(size_t)i * 4 + 3] = s_rpn[b * 5 + 4];
        out[(size_t)N * 4 + i] = s_sc[b * 2 + 1];
    }
}

extern "C" void kernel_launch(void* const* d_in, const int* in_sizes, int n_in,
                              void* d_out, int out_size, void* d_ws, size_t ws_size,
                              hipStream_t stream) {
    const float* rois = (const float*)d_in[0];   // (N,4)
    const float* rpn  = (const float*)d_in[1];   // (N,NP,5)
    const float* sco  = (const float*)d_in[2];   // (N,NP,2)
    float* out = (float*)d_out;

    const int N  = in_sizes[0] / 4;
    const int NP = in_sizes[1] / (N * 5);

    trnms_kernel<<<N, BLK, 0, stream>>>(rois, rpn, sco, out, N, NP);
}